// Encoder_23862838296847
// MI455X (gfx1250) — compile-verified
//
#include <hip/hip_runtime.h>
#include <hip/hip_bf16.h>

typedef float v2f __attribute__((ext_vector_type(2)));
typedef float v8f __attribute__((ext_vector_type(8)));

#define N_NODES 100000
#define E_EDGES 1600000
#define F_IN    256
#define H_HEADS 4
#define D_HID   32
#define HD      128            // H*D
#define NEG_SLOPE 0.2f
#define A_STRIDE 260           // 256 + 4 pad: conflict-free column reads, 16B-aligned rows

// ---------- float -> monotonic uint encoding for atomicMax-based segment max ----------
__device__ __forceinline__ unsigned fenc(float f) {
    unsigned u = __float_as_uint(f);
    return (u & 0x80000000u) ? ~u : (u | 0x80000000u);
}
__device__ __forceinline__ float fdec(unsigned e) {
    return (e & 0x80000000u) ? __uint_as_float(e & 0x7FFFFFFFu) : __uint_as_float(~e);
}
__device__ __forceinline__ float leaky(float v) { return v > 0.0f ? v : NEG_SLOPE * v; }

// ---------- init: zero the output accumulator, set emax=-inf(encoded), denom=0 ----------
__global__ __launch_bounds__(256) void init_kernel(float* __restrict__ agg,
                                                   unsigned* __restrict__ emax,
                                                   float* __restrict__ denom) {
    int i = blockIdx.x * blockDim.x + threadIdx.x;
    if (i < N_NODES * HD) agg[i] = 0.0f;
    if (i < N_NODES * H_HEADS) { emax[i] = fenc(-__builtin_inff()); denom[i] = 0.0f; }
}

// ---------- gathered projection GEMM: h[n,:] = features[perm[n],:] @ W  (fp32 WMMA) ----------
// Block = 256 threads = 8 waves. Block handles a 16-row stripe; wave w owns column tile w (16 cols).
// K loop: 64 steps of V_WMMA_F32_16X16X4_F32.
__global__ __launch_bounds__(256) void gemm_kernel(const float* __restrict__ feat,
                                                   const float* __restrict__ Wm,
                                                   const int*   __restrict__ perm,
                                                   float* __restrict__ h) {
    __shared__ float As[16 * A_STRIDE];
    const int rt = blockIdx.x;            // row tile: rows rt*16 .. rt*16+15
    const int t  = threadIdx.x;

    // Stage gathered A tile (16 x 256 fp32) into LDS. Thread t loads row t/16, 16 cols at (t%16)*16.
    {
        const int i  = t >> 4;            // row within tile
        const int c0 = (t & 15) << 4;     // col start
        const int srow = perm[rt * 16 + i];
        const float4* s = (const float4*)(feat + (size_t)srow * F_IN + c0);
        float4 v0 = s[0], v1 = s[1], v2 = s[2], v3 = s[3];
        float* d = As + i * A_STRIDE + c0;
        *(float4*)(d + 0)  = v0;
        *(float4*)(d + 4)  = v1;
        *(float4*)(d + 8)  = v2;
        *(float4*)(d + 12) = v3;
    }
    __syncthreads();

    const int wave    = t >> 5;           // 0..7 -> column tile
    const int lane    = t & 31;
    const int colbase = wave * 16;
    const int row16   = lane & 15;        // M (A frag) / N (B frag)
    const int khalf   = (lane >> 4) * 2;  // lanes 0-15 hold K=0,1; lanes 16-31 hold K=2,3

    v8f acc = {};
    #pragma unroll 4
    for (int k = 0; k < F_IN; k += 4) {
        v2f a, b;
        const float* ap = As + row16 * A_STRIDE + k + khalf;
        a.x = ap[0];
        a.y = ap[1];
        const float* wp = Wm + (size_t)(k + khalf) * HD + colbase + row16;
        b.x = wp[0];
        b.y = wp[HD];
        acc = __builtin_amdgcn_wmma_f32_16x16x4_f32(false, a, false, b,
                                                    (short)0, acc, false, false);
    }

    // C/D layout: VGPR r -> lanes 0-15 M=r, lanes 16-31 M=8+r; N = lane&15.
    const int col     = colbase + row16;
    const int rowbase = rt * 16 + (lane >> 4) * 8;
    #pragma unroll
    for (int r = 0; r < 8; ++r)
        h[(size_t)(rowbase + r) * HD + col] = acc[r];
}

// ---------- per-node attention half-scores: el/er [N,H] ----------
__global__ __launch_bounds__(256) void attn_half_kernel(const float* __restrict__ h,
                                                        const float* __restrict__ al,
                                                        const float* __restrict__ ar,
                                                        float* __restrict__ el,
                                                        float* __restrict__ er) {
    int i = blockIdx.x * blockDim.x + threadIdx.x;   // node*H + head
    if (i >= N_NODES * H_HEADS) return;
    const int head = i & (H_HEADS - 1);
    const float* hp  = h  + (size_t)(i >> 2) * HD + head * D_HID;
    const float* alp = al + head * D_HID;
    const float* arp = ar + head * D_HID;
    float sl = 0.0f, sr = 0.0f;
    #pragma unroll
    for (int d = 0; d < D_HID; ++d) { float v = hp[d]; sl += v * alp[d]; sr += v * arp[d]; }
    el[i] = sl;
    er[i] = sr;
}

// ---------- edge pass 1: segment max of leaky_relu(el[src]+er[dst]) over dst ----------
__global__ __launch_bounds__(256) void edge_max_kernel(const int* __restrict__ src,
                                                       const int* __restrict__ dst,
                                                       const float* __restrict__ el,
                                                       const float* __restrict__ er,
                                                       unsigned* __restrict__ emax) {
    int e = blockIdx.x * blockDim.x + threadIdx.x;
    if (e >= E_EDGES) return;
    const int s = src[e], d = dst[e];
    #pragma unroll
    for (int hh = 0; hh < H_HEADS; ++hh) {
        float v = leaky(el[s * H_HEADS + hh] + er[d * H_HEADS + hh]);
        atomicMax(&emax[d * H_HEADS + hh], fenc(v));
    }
}

// ---------- edge pass 2: denom[dst] += exp(e - emax[dst]) ----------
__global__ __launch_bounds__(256) void edge_sum_kernel(const int* __restrict__ src,
                                                       const int* __restrict__ dst,
                                                       const float* __restrict__ el,
                                                       const float* __restrict__ er,
                                                       const unsigned* __restrict__ emax,
                                                       float* __restrict__ denom) {
    int e = blockIdx.x * blockDim.x + threadIdx.x;
    if (e >= E_EDGES) return;
    const int s = src[e], d = dst[e];
    #pragma unroll
    for (int hh = 0; hh < H_HEADS; ++hh) {
        float v  = leaky(el[s * H_HEADS + hh] + er[d * H_HEADS + hh]);
        float ee = __expf(v - fdec(emax[d * H_HEADS + hh]));
        atomicAdd(&denom[d * H_HEADS + hh], ee);
    }
}

// ---------- edge pass 3: agg[dst, head, :] += h[src, head, :] * alpha  (one wave per edge-head) ----------
__global__ __launch_bounds__(256) void edge_agg_kernel(const int* __restrict__ src,
                                                       const int* __restrict__ dst,
                                                       const float* __restrict__ el,
                                                       const float* __restrict__ er,
                                                       const unsigned* __restrict__ emax,
                                                       const float* __restrict__ denom,
                                                       const float* __restrict__ h,
                                                       float* __restrict__ agg) {
    const long long gw = ((long long)blockIdx.x * blockDim.x + threadIdx.x) >> 5; // global wave
    const int lane = threadIdx.x & 31;                                            // d index (wave32, D=32)
    if (gw >= (long long)E_EDGES * H_HEADS) return;
    const int e    = (int)(gw >> 2);
    const int head = (int)(gw & 3);
    const int s = src[e], d = dst[e];
    float v     = leaky(el[s * H_HEADS + head] + er[d * H_HEADS + head]);
    float alpha = __expf(v - fdec(emax[d * H_HEADS + head])) / denom[d * H_HEADS + head];
    float m = h[(size_t)s * HD + head * D_HID + lane] * alpha;
    atomicAdd(&agg[(size_t)d * HD + head * D_HID + lane], m);
}

// ---------- finalize: ELU in place on the accumulator (== d_out) ----------
__global__ __launch_bounds__(256) void elu_kernel(float* __restrict__ out) {
    int i = blockIdx.x * blockDim.x + threadIdx.x;
    if (i >= N_NODES * HD) return;
    float x = out[i];
    out[i] = x > 0.0f ? x : (__expf(x) - 1.0f);
}

extern "C" void kernel_launch(void* const* d_in, const int* in_sizes, int n_in,
                              void* d_out, int out_size, void* d_ws, size_t ws_size,
                              hipStream_t stream) {
    const float* feat = (const float*)d_in[0];   // [N, 256]
    const float* Wm   = (const float*)d_in[1];   // [256, 128]
    const float* al   = (const float*)d_in[2];   // [4, 32]
    const float* ar   = (const float*)d_in[3];   // [4, 32]
    const int*   src  = (const int*)d_in[4];     // [E]
    const int*   dst  = (const int*)d_in[5];     // [E]
    const int*   perm = (const int*)d_in[6];     // [N]
    float* out = (float*)d_out;                  // [N, 128] — also used as the aggregation buffer

    // workspace layout (floats): h[N*128] | el[N*4] | er[N*4] | emax[N*4](u32) | denom[N*4]
    float*    h     = (float*)d_ws;
    float*    el    = h  + (size_t)N_NODES * HD;
    float*    er    = el + (size_t)N_NODES * H_HEADS;
    unsigned* emax  = (unsigned*)(er + (size_t)N_NODES * H_HEADS);
    float*    denom = (float*)(emax + (size_t)N_NODES * H_HEADS);

    init_kernel<<<(N_NODES * HD + 255) / 256, 256, 0, stream>>>(out, emax, denom);

    gemm_kernel<<<N_NODES / 16, 256, 0, stream>>>(feat, Wm, perm, h);

    attn_half_kernel<<<(N_NODES * H_HEADS + 255) / 256, 256, 0, stream>>>(h, al, ar, el, er);

    edge_max_kernel<<<(E_EDGES + 255) / 256, 256, 0, stream>>>(src, dst, el, er, emax);
    edge_sum_kernel<<<(E_EDGES + 255) / 256, 256, 0, stream>>>(src, dst, el, er, emax, denom);

    // one wave32 per (edge, head): E*H waves, 8 waves per block
    edge_agg_kernel<<<(E_EDGES * H_HEADS) / 8, 256, 0, stream>>>(src, dst, el, er, emax, denom, h, out);

    elu_kernel<<<(N_NODES * HD + 255) / 256, 256, 0, stream>>>(out);
}